// BackupBarrierCBF_58179626992195
// MI455X (gfx1250) — compile-verified
//
#include <hip/hip_runtime.h>
#include <hip/hip_bf16.h>
#include <stdint.h>

#define N_T   50
#define TPB   256
#define FPT   15              // floats per task record
#define TILE_F (TPB * FPT)    // 3840 floats = 15360 bytes per block tile

typedef uint32_t u32x4 __attribute__((ext_vector_type(4)));
typedef int32_t  i32x4 __attribute__((ext_vector_type(4)));
typedef int32_t  i32x8 __attribute__((ext_vector_type(8)));

// (sigmoid(x) - 0.5) * 2  ==  tanh(x/2), computed stably with one v_exp_f32.
__device__ __forceinline__ float sgm2(float x) {
    float t = __expf(-fabsf(x));
    float r = (1.0f - t) * __builtin_amdgcn_rcpf(1.0f + t);
    return copysignf(r, x);
}

__global__ __launch_bounds__(TPB)
void cbf_kernel(const float* __restrict__ data, float* __restrict__ out, int n) {
    __shared__ float smem[TILE_F];
    const int tid = threadIdx.x;
    const long long base = (long long)blockIdx.x * TILE_F;
    const long long total = (long long)n * FPT;

    // CDNA5 Tensor Data Mover: one wave DMAs this block's 15360B tile into LDS.
    if (tid < 32) {
        uint64_t gaddr    = (uint64_t)(uintptr_t)(data + base);
        uint32_t lds_addr = (uint32_t)(uintptr_t)(void*)&smem[0];   // flat LDS addr[31:0] = LDS byte offset
        long long rem = total - base;
        uint32_t tdim0 = (rem >= (long long)TILE_F) ? (uint32_t)TILE_F
                         : (rem > 0 ? (uint32_t)rem : 1u);          // OOB reads return 0

        // D# group 0: count=1 | lds_addr | global_addr[56:0] | type=2
        u32x4 g0;
        g0.x = 1u;
        g0.y = lds_addr;
        g0.z = (uint32_t)gaddr;
        g0.w = (uint32_t)((gaddr >> 32) & 0x1FFFFFFu) | (2u << 30);

        // D# group 1: data_size=2 (4B), tensor_dim0=tdim0, tensor_dim1=1,
        //             tile_dim0=3840, tile_dim1/2 unused, dim0_stride=3840
        i32x8 g1;
        g1[0] = (int)(2u << 16);                      // wg_mask=0, data_size=4B
        g1[1] = (int)((tdim0 & 0xFFFFu) << 16);       // tensor_dim0[15:0]  @ bits 63:48
        g1[2] = (int)((tdim0 >> 16) | (1u << 16));    // tensor_dim0[31:16], tensor_dim1=1
        g1[3] = (int)((uint32_t)TILE_F << 16);        // tile_dim0 @ bits 127:112
        g1[4] = 0;                                    // tile_dim1=0, tile_dim2=0
        g1[5] = (int)TILE_F;                          // tensor_dim0_stride[31:0]
        g1[6] = 0;
        g1[7] = 0;
        i32x4 g2 = {0, 0, 0, 0};                      // group 2 (unused, <=2-D)
        i32x4 g3 = {0, 0, 0, 0};                      // group 3 (unused)
        i32x8 g4 = {0, 0, 0, 0, 0, 0, 0, 0};          // extra group (clang-23 6-arg form)

        __builtin_amdgcn_tensor_load_to_lds(g0, g1, g2, g3, g4, 0);
        __builtin_amdgcn_s_wait_tensorcnt(0);
    }
    __syncthreads();

    // Per-thread state (records in LDS: stride 15 floats — coprime with 64 banks,
    // so the strided reads are bank-conflict-free).
    const float* p = &smem[tid * FPT];
    float xe = p[0],  ye = p[1],  ve = p[2],  the = p[3];
    float xa = p[4],  ya = p[5],  va = p[6],  tha = p[7];
    float eex = p[8], eey = p[9];
    float aex = p[11], aey = p[12];
    float dt  = p[14];

    // Headings never change (turn == 0): hoist their sin/cos.
    float se0, ce0, sa0, ca0;
    __sincosf(the, &se0, &ce0);
    __sincosf(tha, &sa0, &ca0);

    float r_ag = 0.5f * __fsqrt_rn(aex * aex + aey * aey);
    float r_eg = 0.5f * __fsqrt_rn(eex * eex + eey * eey);
    float off1x = 0.5f * eex + r_ag, off1y = 0.5f * eey + r_ag;
    float off2x = 0.5f * aex + r_eg, off2y = 0.5f * aey + r_eg;

    float h = 3.402823466e38f;
#pragma unroll 2
    for (int t = 0; t < N_T; ++t) {
        // braking_controller on PRE-update state
        float be = -9.0f * sgm2(4.0f * ve);
        float ba = -9.0f * sgm2(4.0f * va);
        // unicycle_step (theta constant)
        xe = fmaf(dt * ve, ce0, xe);
        ye = fmaf(dt * ve, se0, ye);
        xa = fmaf(dt * va, ca0, xa);
        ya = fmaf(dt * va, sa0, ya);
        ve = fmaf(dt, be, ve);
        va = fmaf(dt, ba, va);

        // veh_veh_distance on POST-update state; rotation angle is traj[...,2] == v(t)
        float dx = xa - xe, dy = ya - ye;
        float se, ce, sa, ca;
        __sincosf(ve, &se, &ce);
        __sincosf(va, &sa, &ca);
        // rotate2d(d, -ve) = (ce*dx + se*dy, ce*dy - se*dx); rotate2d(-d, -va) up to sign under abs
        float d1 = fmaxf(fabsf(ce * dx + se * dy) - off1x,
                         fabsf(ce * dy - se * dx) - off1y);
        float d2 = fmaxf(fabsf(ca * dx + sa * dy) - off2x,
                         fabsf(sa * dx - ca * dy) - off2y);
        h = fminf(h, fmaxf(d1, d2));
    }

    int idx = blockIdx.x * TPB + tid;
    if (idx < n) out[idx] = 5.0f * sgm2(h * 0.2f);   // (sigmoid(h/5)-0.5)*2*5
}

extern "C" void kernel_launch(void* const* d_in, const int* in_sizes, int n_in,
                              void* d_out, int out_size, void* d_ws, size_t ws_size,
                              hipStream_t stream) {
    (void)n_in; (void)d_ws; (void)ws_size; (void)out_size;
    const float* data = (const float*)d_in[0];
    float* out = (float*)d_out;
    int n = in_sizes[0] / FPT;                 // B*A tasks (65536)
    int blocks = (n + TPB - 1) / TPB;
    cbf_kernel<<<blocks, TPB, 0, stream>>>(data, out, n);
}